// GCN_45105746542606
// MI455X (gfx1250) — compile-verified
//
#include <hip/hip_runtime.h>

typedef __attribute__((ext_vector_type(2))) float v2f;
typedef __attribute__((ext_vector_type(8))) float v8f;

#define NN 50000
#define NE 800000
#define DD 128

// ---------------------------------------------------------------- transpose
__global__ __launch_bounds__(256) void gcn_transpose(const float* __restrict__ W,
                                                     float* __restrict__ Wt) {
    int idx = blockIdx.x * 256 + threadIdx.x;
    if (idx < DD * DD) {
        int r = idx >> 7, c = idx & 127;
        Wt[c * DD + r] = W[idx];
    }
}

// ---------------------------------------------------------------- degrees
__global__ __launch_bounds__(256) void gcn_deg_init(float* __restrict__ deg) {
    int i = blockIdx.x * 256 + threadIdx.x;
    if (i < NN) deg[i] = 1.0f;  // self loop
}

__global__ __launch_bounds__(256) void gcn_deg_count(const int* __restrict__ col,
                                                     float* __restrict__ deg) {
    int e = blockIdx.x * 256 + threadIdx.x;
    if (e < NE) atomicAdd(&deg[col[e]], 1.0f);
}

__global__ __launch_bounds__(256) void gcn_deg_finish(float* __restrict__ deg) {
    int i = blockIdx.x * 256 + threadIdx.x;
    if (i < NN) deg[i] = rsqrtf(deg[i]);  // deg >= 1 always (self loops)
}

// ---------------------------------------------------------------- WMMA GEMM
// H[nrows x 128] = X[nrows x 128] * W, with Wt = W^T (row n of Wt = col n of W).
// Block: 128 threads = 4 waves; block owns a 16-row strip; wave w owns column
// tiles [32w, 32w+16) and [32w+16, 32w+32).
__global__ __launch_bounds__(128) void gcn_gemm_wmma(const float* __restrict__ X,
                                                     const float* __restrict__ Wt,
                                                     float* __restrict__ H,
                                                     int nrows) {
    __shared__ float As[16][132];  // padded: stride 132 -> conflict-free ds reads
    const int m0   = blockIdx.x * 16;
    const int tid  = threadIdx.x;
    const int wave = tid >> 5;
    const int lane = tid & 31;

    // cooperative load of the 16x128 A strip
    for (int i = tid; i < 16 * DD; i += 128) {
        int r = i >> 7, c = i & 127;
        float v = 0.0f;
        if (m0 + r < nrows) v = X[(size_t)(m0 + r) * DD + c];
        As[r][c] = v;
    }
    __syncthreads();

    const int m  = lane & 15;          // A row owned by this lane
    const int kb = (lane >> 4) << 1;   // K sub-offset: 0 (lanes 0-15) / 2 (16-31)
    const int n  = lane & 15;          // B/D column owned by this lane
    const int n0 = wave * 32;
    const int n1 = n0 + 16;

    const float* wt0 = Wt + (size_t)(n0 + n) * DD;  // contiguous K for col n0+n
    const float* wt1 = Wt + (size_t)(n1 + n) * DD;

    v8f acc0 = {};
    v8f acc1 = {};
    for (int k = 0; k < DD; k += 4) {
        v2f a  = *(const v2f*)&As[m][k + kb];        // A: M=m, K=k+kb..k+kb+1
        v2f b0 = *(const v2f*)(wt0 + k + kb);        // B: K rows k+kb..+1, N=n0+n
        v2f b1 = *(const v2f*)(wt1 + k + kb);
        acc0 = __builtin_amdgcn_wmma_f32_16x16x4_f32(false, a, false, b0,
                                                     (short)0, acc0, false, false);
        acc1 = __builtin_amdgcn_wmma_f32_16x16x4_f32(false, a, false, b1,
                                                     (short)0, acc1, false, false);
    }

    // C/D layout: VGPR r -> M=r (lanes 0-15) / M=8+r (lanes 16-31); N=lane&15
    const int mhalf = (lane >> 4) * 8;
    for (int r = 0; r < 8; ++r) {
        int mm = m0 + mhalf + r;
        if (mm < nrows) {
            H[(size_t)mm * DD + n0 + n] = acc0[r];
            H[(size_t)mm * DD + n1 + n] = acc1[r];
        }
    }
}

// ------------------------------------------------ self-loop + bias (full write)
__global__ __launch_bounds__(256) void gcn_self_init(const float4* __restrict__ H4,
                                                     const float* __restrict__ dinv,
                                                     const float4* __restrict__ b4,
                                                     float4* __restrict__ O4) {
    int idx = blockIdx.x * 256 + threadIdx.x;  // over N*D/4
    if (idx < NN * DD / 4) {
        int i  = idx >> 5;        // node (32 float4 per row)
        int d4 = idx & 31;        // float4 column group
        float s  = dinv[i];
        float ss = s * s;
        float4 h = H4[idx];
        float4 b = b4[d4];
        float4 o;
        o.x = ss * h.x + b.x;
        o.y = ss * h.y + b.y;
        o.z = ss * h.z + b.z;
        o.w = ss * h.w + b.w;
        O4[idx] = o;
    }
}

// ------------------------------------------------ edge scatter (1 wave / edge)
__global__ __launch_bounds__(256) void gcn_edge_scatter(const int* __restrict__ row,
                                                        const int* __restrict__ col,
                                                        const float* __restrict__ dinv,
                                                        const float* __restrict__ H,
                                                        float* __restrict__ out) {
    int e    = (blockIdx.x * 256 + threadIdx.x) >> 5;  // wave index == edge
    int lane = threadIdx.x & 31;
    if (e < NE) {
        int r = row[e];
        int c = col[e];
        float norm = dinv[r] * dinv[c];
        float4 v = ((const float4*)(H + (size_t)r * DD))[lane];
        float* o = out + (size_t)c * DD + lane * 4;
        atomicAdd(o + 0, norm * v.x);
        atomicAdd(o + 1, norm * v.y);
        atomicAdd(o + 2, norm * v.z);
        atomicAdd(o + 3, norm * v.w);
    }
}

// ---------------------------------------------------------------- ReLU
__global__ __launch_bounds__(256) void gcn_relu(float4* __restrict__ a) {
    int idx = blockIdx.x * 256 + threadIdx.x;
    if (idx < NN * DD / 4) {
        float4 v = a[idx];
        v.x = fmaxf(v.x, 0.0f);
        v.y = fmaxf(v.y, 0.0f);
        v.z = fmaxf(v.z, 0.0f);
        v.w = fmaxf(v.w, 0.0f);
        a[idx] = v;
    }
}

// ---------------------------------------------------------------- launcher
extern "C" void kernel_launch(void* const* d_in, const int* in_sizes, int n_in,
                              void* d_out, int out_size, void* d_ws, size_t ws_size,
                              hipStream_t stream) {
    const float* x  = (const float*)d_in[0];
    const int*   ei = (const int*)d_in[1];   // [2, NE]: row then col
    const float* W1 = (const float*)d_in[2];
    const float* b1 = (const float*)d_in[3];
    const float* W2 = (const float*)d_in[4];
    const float* b2 = (const float*)d_in[5];
    const int* row = ei;
    const int* col = ei + NE;
    float* out = (float*)d_out;

    // workspace carve-up (256B aligned slices)
    char* ws = (char*)d_ws;
    size_t off = 0;
    auto carve = [&](size_t bytes) -> float* {
        float* p = (float*)(ws + off);
        off = (off + bytes + 255) & ~(size_t)255;
        return p;
    };
    float* dinv = carve((size_t)NN * 4);
    float* Wt1  = carve((size_t)DD * DD * 4);
    float* Wt2  = carve((size_t)DD * DD * 4);
    float* h    = carve((size_t)NN * DD * 4);
    float* a1   = carve((size_t)NN * DD * 4);

    const int nd4      = NN * DD / 4;       // 1,600,000
    const int nd4_blk  = (nd4 + 255) / 256; // 6250
    const int n_blk    = (NN + 255) / 256;
    const int e_blk    = (NE + 255) / 256;
    const int ew_blk   = (NE * 32 + 255) / 256;  // 1 wave per edge -> 100000 blocks
    const int gemm_blk = (NN + 15) / 16;         // 3125
    const int t_blk    = (DD * DD + 255) / 256;

    // weight transposes
    gcn_transpose<<<t_blk, 256, 0, stream>>>(W1, Wt1);
    gcn_transpose<<<t_blk, 256, 0, stream>>>(W2, Wt2);

    // degree -> dinv
    gcn_deg_init<<<n_blk, 256, 0, stream>>>(dinv);
    gcn_deg_count<<<e_blk, 256, 0, stream>>>(col, dinv);
    gcn_deg_finish<<<n_blk, 256, 0, stream>>>(dinv);

    // ----- layer 1 -----
    gcn_gemm_wmma<<<gemm_blk, 128, 0, stream>>>(x, Wt1, h, NN);
    gcn_self_init<<<nd4_blk, 256, 0, stream>>>((const float4*)h, dinv,
                                               (const float4*)b1, (float4*)a1);
    gcn_edge_scatter<<<ew_blk, 256, 0, stream>>>(row, col, dinv, h, a1);
    gcn_relu<<<nd4_blk, 256, 0, stream>>>((float4*)a1);

    // ----- layer 2 -----
    gcn_gemm_wmma<<<gemm_blk, 128, 0, stream>>>(a1, Wt2, h, NN);
    gcn_self_init<<<nd4_blk, 256, 0, stream>>>((const float4*)h, dinv,
                                               (const float4*)b2, (float4*)out);
    gcn_edge_scatter<<<ew_blk, 256, 0, stream>>>(row, col, dinv, h, out);
    gcn_relu<<<nd4_blk, 256, 0, stream>>>((float4*)out);
}